// BreadthAttentionConv_29188597744226
// MI455X (gfx1250) — compile-verified
//
#include <hip/hip_runtime.h>
#include <hip/hip_bf16.h>

#define NN 100000
#define EE 1600000
#define DD 64
#define EN (EE + NN)
#define EPSF 1e-12f

typedef __attribute__((ext_vector_type(2))) float v2f;
typedef __attribute__((ext_vector_type(8))) float v8f;

// ---- ordered-uint encoding for float atomic max (monotone for all floats) ----
__device__ __forceinline__ unsigned f2ord(float f) {
    unsigned u = __float_as_uint(f);
    return (u & 0x80000000u) ? ~u : (u | 0x80000000u);
}
__device__ __forceinline__ float ord2f(unsigned k) {
    unsigned u = (k & 0x80000000u) ? (k & 0x7FFFFFFFu) : ~k;
    return __uint_as_float(u);
}

__device__ __forceinline__ v8f wmma_f32(v2f a, v2f b, v8f c) {
    // D = A(16x4, f32) x B(4x16, f32) + C(16x16, f32)
    return __builtin_amdgcn_wmma_f32_16x16x4_f32(
        /*neg_a=*/false, a, /*neg_b=*/false, b,
        /*c_mod=*/(short)0, c, /*reuse_a=*/false, /*reuse_b=*/false);
}

// ---------------- init: zero output, -inf segmax (encoded), zero denom -------
__global__ void init_kernel(float* __restrict__ out, unsigned* __restrict__ smax,
                            float* __restrict__ den) {
    int i = blockIdx.x * blockDim.x + threadIdx.x;
    if (i < NN * DD) out[i] = 0.0f;
    if (i < NN) { smax[i] = 0x007FFFFFu; /* f2ord(-inf) */ den[i] = 0.0f; }
}

// ---------------- node projections via WMMA f32 16x16x4 ----------------------
// One wave per 16-node tile; computes hs = h@Ws^T, hd = h@Wd^T, hm = h@Wm^T.
// A[m][k]: lane = m + 16*(k>>1), vgpr = k&1  (fp32 A 16x4 layout)
// B[k][n]: lane = n + 16*(k>>1), vgpr = k&1  (fp32 B 4x16 layout)
// D[m][n]: vgpr r holds M = r + 8*(lane>>4), N = lane&15
__global__ void __launch_bounds__(32) proj_kernel(
    const float* __restrict__ h,
    const float* __restrict__ Ws, const float* __restrict__ Wd,
    const float* __restrict__ Wm,
    float* __restrict__ hs, float* __restrict__ hd, float* __restrict__ hm) {
    const int tile = blockIdx.x;          // 16 nodes per tile
    const int lane = threadIdx.x;         // 0..31
    const int m    = lane & 15;
    const int kh   = lane >> 4;           // which K half-pair (adds 2 to k)

    v8f accS[4] = {}; v8f accD[4] = {}; v8f accM[4] = {};

    // Preload all A fragments for this 16-node tile (one clause of 16 b64 loads)
    const size_t rowA = (size_t)(tile * 16 + m) * DD;
    v2f a[16];
#pragma unroll
    for (int k = 0; k < 16; ++k)
        a[k] = *(const v2f*)(h + rowA + 4 * k + 2 * kh);

    // Fully unrolled MAC loop: lets the scheduler pipeline the 192 L2-resident
    // weight-fragment loads against the 192 WMMAs instead of stalling per load.
#pragma unroll
    for (int k = 0; k < 16; ++k) {
#pragma unroll
        for (int nt = 0; nt < 4; ++nt) {
            const int off = (nt * 16 + m) * DD + 4 * k + 2 * kh;
            accS[nt] = wmma_f32(a[k], *(const v2f*)(Ws + off), accS[nt]);
            accD[nt] = wmma_f32(a[k], *(const v2f*)(Wd + off), accD[nt]);
            accM[nt] = wmma_f32(a[k], *(const v2f*)(Wm + off), accM[nt]);
        }
    }

#pragma unroll
    for (int nt = 0; nt < 4; ++nt) {
#pragma unroll
        for (int r = 0; r < 8; ++r) {
            const size_t row = (size_t)(tile * 16 + r + 8 * kh);
            const int    col = nt * 16 + m;
            hs[row * DD + col] = accS[nt][r];
            hd[row * DD + col] = accD[nt][r];
            hm[row * DD + col] = accM[nt][r];
        }
    }
}

// ---------------- edge scores: e = v . tanh(hs[dst] + hd[src]) ---------------
__global__ void score_kernel(const float* __restrict__ hs, const float* __restrict__ hd,
                             const int* __restrict__ ei, const float* __restrict__ v,
                             float* __restrict__ e, unsigned* __restrict__ smax) {
    __shared__ float vs[DD];
    if (threadIdx.x < DD) vs[threadIdx.x] = v[threadIdx.x];
    __syncthreads();
    const int i = blockIdx.x * blockDim.x + threadIdx.x;
    if (i >= EN) return;
    const int src = (i < EE) ? ei[i]      : (i - EE);
    const int dst = (i < EE) ? ei[EE + i] : (i - EE);
    const float4* a4 = (const float4*)(hs + (size_t)dst * DD);
    const float4* b4 = (const float4*)(hd + (size_t)src * DD);
    float acc = 0.0f;
#pragma unroll
    for (int k = 0; k < DD / 4; ++k) {
        float4 a = a4[k], b = b4[k];
        acc += tanhf(a.x + b.x) * vs[4 * k + 0];
        acc += tanhf(a.y + b.y) * vs[4 * k + 1];
        acc += tanhf(a.z + b.z) * vs[4 * k + 2];
        acc += tanhf(a.w + b.w) * vs[4 * k + 3];
    }
    e[i] = acc;
    atomicMax(&smax[dst], f2ord(acc));
}

// ---------------- exp(e - segmax) and denom accumulation ---------------------
__global__ void exp_kernel(const int* __restrict__ ei, const unsigned* __restrict__ smax,
                           float* __restrict__ e, float* __restrict__ den) {
    const int i = blockIdx.x * blockDim.x + threadIdx.x;
    if (i >= EN) return;
    const int dst = (i < EE) ? ei[EE + i] : (i - EE);
    const float mx = ord2f(smax[dst]);
    const float ex = __expf(e[i] - mx);
    e[i] = ex;
    atomicAdd(&den[dst], ex);
}

// ---------------- weighted scatter-sum, wave-coalesced atomics ---------------
// Each lane owns one edge; the wave processes edges one at a time with lanes
// covering dims [lane] and [lane+32] -> coalesced atomic adds.
__global__ void scatter_kernel(const int* __restrict__ ei, const float* __restrict__ e,
                               const float* __restrict__ den, const float* __restrict__ hm,
                               float* __restrict__ out) {
    const int i    = blockIdx.x * blockDim.x + threadIdx.x;
    const int lane = threadIdx.x & 31;
    int src = 0, dst = 0;
    float alpha = 0.0f;
    if (i < EN) {
        src = (i < EE) ? ei[i]      : (i - EE);
        dst = (i < EE) ? ei[EE + i] : (i - EE);
        alpha = e[i] / fmaxf(den[dst], EPSF);
    }
#pragma unroll 1
    for (int j = 0; j < 32; ++j) {
        const int   ej = __shfl(i, j, 32);
        const int   s  = __shfl(src, j, 32);
        const int   d  = __shfl(dst, j, 32);
        const float al = __shfl(alpha, j, 32);
        if (ej < EN) {
            const float m0 = hm[(size_t)s * DD + lane];
            const float m1 = hm[(size_t)s * DD + 32 + lane];
            atomicAdd(&out[(size_t)d * DD + lane],      al * m0);
            atomicAdd(&out[(size_t)d * DD + 32 + lane], al * m1);
        }
    }
}

// ---------------- final tanh -------------------------------------------------
__global__ void finish_kernel(float* __restrict__ out) {
    const int i = blockIdx.x * blockDim.x + threadIdx.x;
    if (i < NN * DD) out[i] = tanhf(out[i]);
}

extern "C" void kernel_launch(void* const* d_in, const int* in_sizes, int n_in,
                              void* d_out, int out_size, void* d_ws, size_t ws_size,
                              hipStream_t stream) {
    const float* h  = (const float*)d_in[0];
    const int*   ei = (const int*)d_in[1];   // (2, E) flat: [0..E) = src, [E..2E) = dst
    const float* Wm = (const float*)d_in[2];
    const float* Ws = (const float*)d_in[3];
    const float* Wd = (const float*)d_in[4];
    const float* v  = (const float*)d_in[5];
    float* out = (float*)d_out;

    float* ws = (float*)d_ws;
    float* hs = ws;
    float* hd = hs + (size_t)NN * DD;
    float* hm = hd + (size_t)NN * DD;
    float* e  = hm + (size_t)NN * DD;
    unsigned* smax = (unsigned*)(e + (size_t)EN);
    float*    den  = (float*)(smax + NN);

    const int tpb = 256;
    init_kernel<<<(NN * DD + tpb - 1) / tpb, tpb, 0, stream>>>(out, smax, den);
    proj_kernel<<<NN / 16, 32, 0, stream>>>(h, Ws, Wd, Wm, hs, hd, hm);
    score_kernel<<<(EN + tpb - 1) / tpb, tpb, 0, stream>>>(hs, hd, ei, v, e, smax);
    exp_kernel<<<(EN + tpb - 1) / tpb, tpb, 0, stream>>>(ei, smax, e, den);
    scatter_kernel<<<(EN + tpb - 1) / tpb, tpb, 0, stream>>>(ei, e, den, hm, out);
    finish_kernel<<<(NN * DD + tpb - 1) / tpb, tpb, 0, stream>>>(out);
}